// KnowledgeBase_20134806683883
// MI455X (gfx1250) — compile-verified
//
#include <hip/hip_runtime.h>

// CDNA5 / gfx1250: wave32, WMMA (not MFMA).
// Full-precision path: V_WMMA_F32_16X16X4_F32, D = A(16x4 f32) x B(4x16 f32) + C(16x16 f32).
typedef __attribute__((ext_vector_type(2))) float v2f;
typedef __attribute__((ext_vector_type(8))) float v8f;

#define KB_Q   3
#define KB_R   128
#define KB_V   64
#define KB_EPS 0.4999999f

// Single-wave kernel. The reference's `% R` on the flattened index means only
// rows 0..127 of storage.reshape(R^3, V) are ever touched (2 KB of data), so
// this is a latency-bound problem: one wave32 does everything.
//
// GEMV out[64] = values[8x64]^T @ w[8] / 3 is mapped onto the matrix pipe:
//   A[m][k] = w[jbase + k]                 (broadcast over M)
//   B[k][n] = values[jbase + k][16*c + n]
//   D[m][n] = sum_j w[j] * values[j][16*c + n]   (all 16 rows identical)
// Two chained K=4 WMMAs cover the 8 neighbors; 4 column-chunks cover V=64.
__global__ void KnowledgeBase_20134806683883_kernel(const float* __restrict__ query,
                                                    const float* __restrict__ storage,
                                                    float* __restrict__ out) {
    const int lane = threadIdx.x & 31;
    const int half = lane >> 4;   // 0: lanes 0-15, 1: lanes 16-31
    const int col  = lane & 15;   // N (and M) coordinate

    // saw[q] = frac(query[q]) * R   (uniform across lanes)
    float saw[KB_Q];
#pragma unroll
    for (int q = 0; q < KB_Q; ++q) {
        float v = query[q];
        saw[q] = (v - floorf(v)) * (float)KB_R;
    }

    // Neighbor map row n, component q: sign = ((n >> (Q-1-q)) & 1) ? -EPS : +EPS.
    // flat_conv = {R^2, R^1, R^0}; faithful Python-sign mod by R afterwards.
    int   flat[8];
    float w[8];
    const int conv[KB_Q] = { KB_R * KB_R, KB_R, 1 };
#pragma unroll
    for (int n = 0; n < 8; ++n) {
        float wsum = 0.0f;
        int   fi   = 0;
#pragma unroll
        for (int q = 0; q < KB_Q; ++q) {
            float off = ((n >> (KB_Q - 1 - q)) & 1) ? -KB_EPS : KB_EPS;
            float idx = rintf(off + saw[q]);      // RNE == jnp.round
            wsum += 1.0f - fabsf(idx - saw[q]);
            fi   += (int)idx * conv[q];           // int32 truncation, exact here
        }
        fi %= KB_R;
        if (fi < 0) fi += KB_R;                   // Python/JAX mod semantics
        flat[n] = fi;
        w[n]    = wsum;
    }

    // A operands: A[m][k] = w[jbase + k]; per ISA layout a.x -> K = 2*half, a.y -> K = 2*half+1.
    v2f a_lo, a_hi;
    a_lo.x = w[2 * half + 0];
    a_lo.y = w[2 * half + 1];
    a_hi.x = w[4 + 2 * half + 0];
    a_hi.y = w[4 + 2 * half + 1];

    // Four 16-wide output chunks; keep all WMMAs in uniform control flow (EXEC all ones).
    float res[4];
#pragma unroll
    for (int c = 0; c < 4; ++c) {
        const int vcol = 16 * c + col;            // value index within row
        // B operands: b.x -> K = 2*half, b.y -> K = 2*half+1, N = col.
        v2f b_lo, b_hi;
        b_lo.x = storage[flat[2 * half + 0] * KB_V + vcol];
        b_lo.y = storage[flat[2 * half + 1] * KB_V + vcol];
        b_hi.x = storage[flat[4 + 2 * half + 0] * KB_V + vcol];
        b_hi.y = storage[flat[4 + 2 * half + 1] * KB_V + vcol];

        v8f acc = {};
        // 8 args: (neg_a, A, neg_b, B, c_mod, C, reuse_a, reuse_b)
        acc = __builtin_amdgcn_wmma_f32_16x16x4_f32(false, a_lo, false, b_lo,
                                                    (short)0, acc, false, false);
        acc = __builtin_amdgcn_wmma_f32_16x16x4_f32(false, a_hi, false, b_hi,
                                                    (short)0, acc, false, false);
        res[c] = acc[0];                          // D[row][col], rows identical
    }

    // Store after all WMMAs (divergence is now safe). Lanes 0-15 cover N=0..15.
    if (lane < 16) {
#pragma unroll
        for (int c = 0; c < 4; ++c) {
            out[16 * c + col] = res[c] / 3.0f;    // divide by Q, faithful to reference
        }
    }
}

extern "C" void kernel_launch(void* const* d_in, const int* in_sizes, int n_in,
                              void* d_out, int out_size, void* d_ws, size_t ws_size,
                              hipStream_t stream) {
    (void)in_sizes; (void)n_in; (void)d_ws; (void)ws_size; (void)out_size;
    const float* query   = (const float*)d_in[0];   // [3] f32
    const float* storage = (const float*)d_in[1];   // [128,128,128,64] f32
    float*       out     = (float*)d_out;           // [64] f32

    // One wave32: the entire problem is 2 KB of reads + 1 KFLOP.
    KnowledgeBase_20134806683883_kernel<<<1, 32, 0, stream>>>(query, storage, out);
}